// TgGAT_53523882443262
// MI455X (gfx1250) — compile-verified
//
#include <hip/hip_runtime.h>
#include <stdint.h>

typedef float v2f __attribute__((ext_vector_type(2)));
typedef float v8f __attribute__((ext_vector_type(8)));

#define DIM 64
#define NEG_SLOPE 0.2f
#define GAT_EPS 1e-16f
#define LDS_STRIDE 80   // DIM + 16: rows k and k+2 hit disjoint LDS bank sets

// ---------------------------------------------------------------------------
// WMMA fp32 GEMM: C[M,64] = A[M,K] @ B[K,64] (+ optional bias[64])
// One wave -> one 16x16 tile via V_WMMA_F32_16X16X4_F32, K fully unrolled.
// Block = 256 threads = 8 waves = 32 rows x 64 cols of C.
// B staged in LDS once per block (conflict-free padded layout).
// ---------------------------------------------------------------------------
template <int K>
__global__ void __launch_bounds__(256)
gemm_wmma_f32_kernel(const float* __restrict__ A, const float* __restrict__ B,
                     const float* __restrict__ bias, float* __restrict__ C,
                     int M) {
    __shared__ float Bs[K * LDS_STRIDE];

    // cooperative stage of B[K,64] into padded LDS (float4 per thread-step)
    #pragma unroll
    for (int i = threadIdx.x; i < K * 16; i += 256) {
        const int row = i >> 4, c4 = (i & 15) << 2;
        const float4 v = ((const float4*)B)[i];
        *(float4*)(&Bs[row * LDS_STRIDE + c4]) = v;
    }
    __syncthreads();

    const int wave = threadIdx.x >> 5;          // 0..7
    const int lane = threadIdx.x & 31;
    const int half = lane >> 4;                 // 0: lanes 0-15, 1: lanes 16-31
    const int l16  = lane & 15;
    const int rowBase = ((blockIdx.x << 1) + (wave >> 2)) << 4;
    const int colBase = (wave & 3) << 4;
    if (rowBase >= M) return;

    const int kLane = 2 * half;                 // lanes 16-31 hold K+2,K+3
    const int col   = colBase + l16;
    v8f acc = {};

    if (rowBase + 16 <= M) {
        // ---- fast path: full tile, no per-iteration guards ----
        const float* ap = A + (size_t)(rowBase + l16) * K + kLane;
        #pragma unroll
        for (int k0 = 0; k0 < K; k0 += 4) {
            const v2f a = *(const v2f*)(ap + k0);
            v2f b;
            b.x = Bs[(k0 + kLane) * LDS_STRIDE + col];
            b.y = Bs[(k0 + kLane + 1) * LDS_STRIDE + col];
            acc = __builtin_amdgcn_wmma_f32_16x16x4_f32(
                false, a, false, b, (short)0, acc, false, false);
        }
    } else {
        // ---- tail path: guard rows ----
        const int r = rowBase + l16;
        const bool rv = (r < M);
        const float* ap = A + (size_t)(rv ? r : 0) * K + kLane;
        #pragma unroll
        for (int k0 = 0; k0 < K; k0 += 4) {
            const v2f t = *(const v2f*)(ap + k0);
            v2f a;
            a.x = rv ? t.x : 0.0f;
            a.y = rv ? t.y : 0.0f;
            v2f b;
            b.x = Bs[(k0 + kLane) * LDS_STRIDE + col];
            b.y = Bs[(k0 + kLane + 1) * LDS_STRIDE + col];
            acc = __builtin_amdgcn_wmma_f32_16x16x4_f32(
                false, a, false, b, (short)0, acc, false, false);
        }
    }

    // C/D layout: VGPR v, lanes 0-15 -> M=v, lanes 16-31 -> M=v+8; N = l16
    const float bv = bias ? bias[col] : 0.0f;
    if (rowBase + 16 <= M) {
        #pragma unroll
        for (int v = 0; v < 8; ++v)
            C[(size_t)(rowBase + v + 8 * half) * DIM + col] = acc[v] + bv;
    } else {
        #pragma unroll
        for (int v = 0; v < 8; ++v) {
            const int r = rowBase + v + 8 * half;
            if (r < M) C[(size_t)r * DIM + col] = acc[v] + bv;
        }
    }
}

// ---------------------------------------------------------------------------
// Attention scores: a_s[n] = h[n,:] . att_src ; a_d[n] = h[n,:] . att_dst
// One wave per node, contiguous float2 per lane, shfl_xor tree reduction.
// ---------------------------------------------------------------------------
__global__ void __launch_bounds__(256)
attn_kernel(const float* __restrict__ h, const float* __restrict__ att_s,
            const float* __restrict__ att_d, float* __restrict__ a_s,
            float* __restrict__ a_d, int N) {
    const int node = (int)((blockIdx.x * blockDim.x + threadIdx.x) >> 5);
    const int lane = threadIdx.x & 31;
    if (node >= N) return;
    const float2 hv = *(const float2*)(h + (size_t)node * DIM + 2 * lane);
    const float2 cs = *(const float2*)(att_s + 2 * lane);
    const float2 cd = *(const float2*)(att_d + 2 * lane);
    float ss = hv.x * cs.x + hv.y * cs.y;
    float sd = hv.x * cd.x + hv.y * cd.y;
    #pragma unroll
    for (int off = 16; off > 0; off >>= 1) {
        ss += __shfl_xor(ss, off, 32);
        sd += __shfl_xor(sd, off, 32);
    }
    if (lane == 0) { a_s[node] = ss; a_d[node] = sd; }
}

// sign-aware float atomic max (memory initialized to -inf)
__device__ __forceinline__ void atomicMaxF32(float* addr, float val) {
    if (val >= 0.0f) atomicMax((int*)addr, __float_as_int(val));
    else             atomicMin((unsigned int*)addr, __float_as_uint(val));
}

__device__ __forceinline__ void edge_sd(const long long* __restrict__ ei,
                                        int e, int E, int& s, int& d) {
    if (e < E) { s = (int)ei[e]; d = (int)ei[E + e]; }
    else       { s = d = e - E; }                      // self-loop tail
}

// pass 1: m[d] = max over incoming edges of leaky_relu(a_s[s]+a_d[d])
__global__ void __launch_bounds__(256)
edge_max_kernel(const long long* __restrict__ ei, const float* __restrict__ a_s,
                const float* __restrict__ a_d, float* __restrict__ m,
                int E, int Etot) {
    const int e = blockIdx.x * blockDim.x + threadIdx.x;
    if (e >= Etot) return;
    int s, d; edge_sd(ei, e, E, s, d);
    float v = a_s[s] + a_d[d];
    v = (v > 0.0f) ? v : NEG_SLOPE * v;
    atomicMaxF32(m + d, v);
}

// pass 2: p[e] = exp(e - m[d]); denom[d] += p[e]
__global__ void __launch_bounds__(256)
edge_exp_kernel(const long long* __restrict__ ei, const float* __restrict__ a_s,
                const float* __restrict__ a_d, const float* __restrict__ m,
                float* __restrict__ p, float* __restrict__ denom,
                int E, int Etot) {
    const int e = blockIdx.x * blockDim.x + threadIdx.x;
    if (e >= Etot) return;
    int s, d; edge_sd(ei, e, E, s, d);
    float v = a_s[s] + a_d[d];
    v = (v > 0.0f) ? v : NEG_SLOPE * v;
    const float pe = __expf(v - m[d]);
    p[e] = pe;
    atomicAdd(denom + d, pe);
}

// pass 3: out[d,:] += (p[e]/(denom[d]+eps)) * h[s,:]   (wave per edge)
__global__ void __launch_bounds__(256)
edge_agg_kernel(const long long* __restrict__ ei, const float* __restrict__ p,
                const float* __restrict__ denom, const float* __restrict__ h,
                float* __restrict__ out, int E, int Etot) {
    const int e = (int)((blockIdx.x * blockDim.x + threadIdx.x) >> 5);
    const int lane = threadIdx.x & 31;
    if (e >= Etot) return;
    int s, d; edge_sd(ei, e, E, s, d);
    const float alpha = p[e] / (denom[d] + GAT_EPS);
    const float2 hv = *(const float2*)(h + (size_t)s * DIM + 2 * lane);
    float* od = out + (size_t)d * DIM + 2 * lane;
    atomicAdd(od,     alpha * hv.x);
    atomicAdd(od + 1, alpha * hv.y);
}

__global__ void fill_kernel(float* __restrict__ p, float v, int n) {
    const int i = blockIdx.x * blockDim.x + threadIdx.x;
    if (i < n) p[i] = v;
}

__global__ void relu_bias_kernel(const float* __restrict__ a,
                                 const float* __restrict__ bias,
                                 float* __restrict__ o, int n) {
    const int i = blockIdx.x * blockDim.x + threadIdx.x;
    if (i < n) {
        const float v = a[i] + bias[i & (DIM - 1)];
        o[i] = v > 0.0f ? v : 0.0f;
    }
}

__global__ void bias_add_kernel(float* __restrict__ o,
                                const float* __restrict__ bias, int n) {
    const int i = blockIdx.x * blockDim.x + threadIdx.x;
    if (i < n) o[i] += bias[i & (DIM - 1)];
}

// ---------------------------------------------------------------------------
extern "C" void kernel_launch(void* const* d_in, const int* in_sizes, int n_in,
                              void* d_out, int out_size, void* d_ws, size_t ws_size,
                              hipStream_t stream) {
    const float*     x    = (const float*)d_in[0];
    const long long* ei   = (const long long*)d_in[1];   // int64 [2,E] flat
    const float*     Wpre = (const float*)d_in[2];
    const float*     bpre = (const float*)d_in[3];
    const float*     W1   = (const float*)d_in[4];
    const float*     as1  = (const float*)d_in[5];
    const float*     ad1  = (const float*)d_in[6];
    const float*     b1   = (const float*)d_in[7];
    const float*     W2   = (const float*)d_in[8];
    const float*     as2  = (const float*)d_in[9];
    const float*     ad2  = (const float*)d_in[10];
    const float*     b2   = (const float*)d_in[11];

    const int IN_DIM = 128;
    const int N    = in_sizes[0] / IN_DIM;
    const int E    = in_sizes[1] / 2;
    const int Etot = E + N;
    const int NF   = N * DIM;

    // workspace carve-out (all fits easily; everything stays L2-resident)
    float* ws    = (float*)d_ws;
    float* h_pre = ws; ws += (size_t)NF;   // reused as h_relu after conv1
    float* h1    = ws; ws += (size_t)NF;   // reused as h2 in conv2
    float* agg   = ws; ws += (size_t)NF;
    float* a_s   = ws; ws += N;
    float* a_d   = ws; ws += N;
    float* m     = ws; ws += N;
    float* denom = ws; ws += N;
    float* p     = ws; ws += Etot;
    float* out   = (float*)d_out;

    const dim3 blk(256);
    auto cdiv = [](int a, int b) { return (a + b - 1) / b; };
    const float ninf = -__builtin_huge_valf();

    // ---- linear_pre + conv1 ----
    fill_kernel<<<cdiv(NF, 256), blk, 0, stream>>>(agg, 0.0f, NF);
    fill_kernel<<<cdiv(N, 256), blk, 0, stream>>>(m, ninf, N);
    fill_kernel<<<cdiv(N, 256), blk, 0, stream>>>(denom, 0.0f, N);

    gemm_wmma_f32_kernel<128><<<cdiv(N, 32), blk, 0, stream>>>(x, Wpre, bpre, h_pre, N);
    gemm_wmma_f32_kernel<64><<<cdiv(N, 32), blk, 0, stream>>>(h_pre, W1, nullptr, h1, N);
    attn_kernel<<<cdiv(N, 8), blk, 0, stream>>>(h1, as1, ad1, a_s, a_d, N);
    edge_max_kernel<<<cdiv(Etot, 256), blk, 0, stream>>>(ei, a_s, a_d, m, E, Etot);
    edge_exp_kernel<<<cdiv(Etot, 256), blk, 0, stream>>>(ei, a_s, a_d, m, p, denom, E, Etot);
    edge_agg_kernel<<<cdiv(Etot, 8), blk, 0, stream>>>(ei, p, denom, h1, agg, E, Etot);
    relu_bias_kernel<<<cdiv(NF, 256), blk, 0, stream>>>(agg, b1, h_pre, NF);  // h_relu

    // ---- conv2 ----
    fill_kernel<<<cdiv(NF, 256), blk, 0, stream>>>(out, 0.0f, NF);
    fill_kernel<<<cdiv(N, 256), blk, 0, stream>>>(m, ninf, N);
    fill_kernel<<<cdiv(N, 256), blk, 0, stream>>>(denom, 0.0f, N);

    gemm_wmma_f32_kernel<64><<<cdiv(N, 32), blk, 0, stream>>>(h_pre, W2, nullptr, h1, N);
    attn_kernel<<<cdiv(N, 8), blk, 0, stream>>>(h1, as2, ad2, a_s, a_d, N);
    edge_max_kernel<<<cdiv(Etot, 256), blk, 0, stream>>>(ei, a_s, a_d, m, E, Etot);
    edge_exp_kernel<<<cdiv(Etot, 256), blk, 0, stream>>>(ei, a_s, a_d, m, p, denom, E, Etot);
    edge_agg_kernel<<<cdiv(Etot, 8), blk, 0, stream>>>(ei, p, denom, h1, out, E, Etot);
    bias_add_kernel<<<cdiv(NF, 256), blk, 0, stream>>>(out, b2, NF);
}